// NonLocalLayer_dilation_75960791597922
// MI455X (gfx1250) — compile-verified
//
#include <hip/hip_runtime.h>
#include <math.h>

typedef __attribute__((ext_vector_type(2))) float v2f;
typedef __attribute__((ext_vector_type(8))) float v8f;

#define CDIM 64
#define HDIM 128
#define WDIM 128
#define PDIM (HDIM * WDIM)   // 16384 pixels per image
#define BDIM 2
#define NPIX (BDIM * PDIM)   // 32768 total pixels

// -----------------------------------------------------------------------------
// Fused QKV GEMM: for one 16(outCh) x 16(pixel) tile, load each x B-fragment
// once and feed three V_WMMA_F32_16X16X4_F32 chains (Wk, Wq, Wv).
// K/Q/V = relu(W* @ x), all fp32.
// -----------------------------------------------------------------------------
__global__ void __launch_bounds__(256) qkv_gemm_wmma(
    const float* __restrict__ Wk, const float* __restrict__ Wq,
    const float* __restrict__ Wv, const float* __restrict__ X,
    float* __restrict__ Ko, float* __restrict__ Qo, float* __restrict__ Vo)
{
  const int tid  = threadIdx.x;
  const int wave = tid >> 5;          // 8 waves per block
  const int lane = tid & 31;
  const int half = lane >> 4;         // 0: lanes 0-15, 1: lanes 16-31
  const int l    = lane & 15;

  const int nTile = (blockIdx.x * 8 + wave) * 16;   // pixel tile
  const int mTile = blockIdx.y * 16;                // out-channel tile

  const int n = nTile + l;            // this lane's pixel (N = l in tile)
  const int b = n >> 14;              // / PDIM
  const int p = n & (PDIM - 1);

  const float* xcol  = X + ((size_t)b * CDIM) * PDIM + p;   // stride PDIM per ch
  const size_t wbase = (size_t)(mTile + l) * CDIM;          // row M = mTile + l
  const float* wkrow = Wk + wbase;
  const float* wqrow = Wq + wbase;
  const float* wvrow = Wv + wbase;

  v8f ck = {0.f,0.f,0.f,0.f,0.f,0.f,0.f,0.f};
  v8f cq = {0.f,0.f,0.f,0.f,0.f,0.f,0.f,0.f};
  v8f cv = {0.f,0.f,0.f,0.f,0.f,0.f,0.f,0.f};

  for (int kk = 0; kk < 16; ++kk) {
    const int kb = kk * 4 + 2 * half;   // this lane's K pair within the 4-chunk
    v2f bf, ak, aq, av;
    bf.x = xcol[(size_t)kb * PDIM];     // B: 4x16, N = l, K = kb, kb+1
    bf.y = xcol[(size_t)(kb + 1) * PDIM];
    ak.x = wkrow[kb]; ak.y = wkrow[kb + 1];
    aq.x = wqrow[kb]; aq.y = wqrow[kb + 1];
    av.x = wvrow[kb]; av.y = wvrow[kb + 1];
    ck = __builtin_amdgcn_wmma_f32_16x16x4_f32(false, ak, false, bf, (short)0, ck, false, false);
    cq = __builtin_amdgcn_wmma_f32_16x16x4_f32(false, aq, false, bf, (short)0, cq, false, false);
    cv = __builtin_amdgcn_wmma_f32_16x16x4_f32(false, av, false, bf, (short)0, cv, false, false);
  }

  const size_t obase = ((size_t)b * CDIM) * PDIM + p;
#pragma unroll
  for (int r = 0; r < 8; ++r) {
    const size_t oofs = obase + (size_t)(mTile + r + 8 * half) * PDIM;
    Ko[oofs] = fmaxf(ck[r], 0.f);
    Qo[oofs] = fmaxf(cq[r], 0.f);
    Vo[oofs] = fmaxf(cv[r], 0.f);
  }
}

// -----------------------------------------------------------------------------
// Output projection GEMM: Y = Wt @ A + residual(x). Same WMMA tiling.
// -----------------------------------------------------------------------------
__global__ void __launch_bounds__(256) proj_gemm_wmma(
    const float* __restrict__ Wm, const float* __restrict__ X,
    float* __restrict__ Y, const float* __restrict__ R)
{
  const int tid  = threadIdx.x;
  const int wave = tid >> 5;
  const int lane = tid & 31;
  const int half = lane >> 4;
  const int l    = lane & 15;

  const int nTile = (blockIdx.x * 8 + wave) * 16;
  const int mTile = blockIdx.y * 16;

  const int n = nTile + l;
  const int b = n >> 14;
  const int p = n & (PDIM - 1);

  const float* xcol = X + ((size_t)b * CDIM) * PDIM + p;
  const float* wrow = Wm + (size_t)(mTile + l) * CDIM;

  v8f acc = {0.f,0.f,0.f,0.f,0.f,0.f,0.f,0.f};

  for (int kk = 0; kk < 16; ++kk) {
    const int kb = kk * 4 + 2 * half;
    v2f a, bf;
    a.x  = wrow[kb];
    a.y  = wrow[kb + 1];
    bf.x = xcol[(size_t)kb * PDIM];
    bf.y = xcol[(size_t)(kb + 1) * PDIM];
    acc = __builtin_amdgcn_wmma_f32_16x16x4_f32(false, a, false, bf, (short)0, acc, false, false);
  }

  const size_t obase = ((size_t)b * CDIM) * PDIM + p;
#pragma unroll
  for (int r = 0; r < 8; ++r) {
    const size_t oofs = obase + (size_t)(mTile + r + 8 * half) * PDIM;
    Y[oofs] = acc[r] + R[oofs];
  }
}

// -----------------------------------------------------------------------------
// Attention over 49 dilated neighbors (k=7, dil=2, replicate pad == clamp).
// One thread per pixel; 49 scores kept in registers; softmax in registers.
// -----------------------------------------------------------------------------
__global__ void __launch_bounds__(256) attn49(
    const float* __restrict__ Kb, const float* __restrict__ Qb,
    const float* __restrict__ Vb, float* __restrict__ Ob)
{
  const int gid = blockIdx.x * 256 + threadIdx.x;
  const int b = gid >> 14;             // / PDIM
  const int p = gid & (PDIM - 1);
  const int h = p >> 7;                // / WDIM
  const int w = p & (WDIM - 1);

  // Precompute clamped neighbor pixel offsets (replicate padding == clamp).
  int np[49];
#pragma unroll
  for (int t = 0; t < 49; ++t) {
    const int dy = (t / 7) * 2 - 6;
    const int dx = (t % 7) * 2 - 6;
    int hc = h + dy; hc = hc < 0 ? 0 : (hc > HDIM - 1 ? HDIM - 1 : hc);
    int wc = w + dx; wc = wc < 0 ? 0 : (wc > WDIM - 1 ? WDIM - 1 : wc);
    np[t] = hc * WDIM + wc;
  }

  const size_t base = (size_t)b * CDIM * PDIM;
  const float* Kp = Kb + base;
  const float* Qp = Qb + base;
  const float* Vp = Vb + base;

  float sc[49];
#pragma unroll
  for (int t = 0; t < 49; ++t) sc[t] = 0.f;

  // Scores: sc[t] = sum_c K[c][neighbor_t] * Q[c][p]
  for (int c = 0; c < CDIM; ++c) {
    const float* Kc = Kp + (size_t)c * PDIM;
    const float  q  = Qp[(size_t)c * PDIM + p];
#pragma unroll
    for (int t = 0; t < 49; ++t)
      sc[t] = fmaf(Kc[np[t]], q, sc[t]);
  }

  // Softmax over the 49 taps.
  float m = sc[0];
#pragma unroll
  for (int t = 1; t < 49; ++t) m = fmaxf(m, sc[t]);
  float s = 0.f;
#pragma unroll
  for (int t = 0; t < 49; ++t) { sc[t] = __expf(sc[t] - m); s += sc[t]; }
  const float inv = 1.f / s;
#pragma unroll
  for (int t = 0; t < 49; ++t) sc[t] *= inv;

  // Aggregation: O[c][p] = sum_t V[c][neighbor_t] * a[t]
  float* Op = Ob + base;
  for (int c = 0; c < CDIM; ++c) {
    const float* Vc = Vp + (size_t)c * PDIM;
    float acc = 0.f;
#pragma unroll
    for (int t = 0; t < 49; ++t)
      acc = fmaf(Vc[np[t]], sc[t], acc);
    Op[(size_t)c * PDIM + p] = acc;
  }
}

// -----------------------------------------------------------------------------
// Launch: fused K/Q/V = relu(W* x)  [1 WMMA kernel, x read once]
//         softmax-attention over 49 taps                    [VALU]
//         out = Wt attnOut + x                              [WMMA + residual]
// -----------------------------------------------------------------------------
extern "C" void kernel_launch(void* const* d_in, const int* in_sizes, int n_in,
                              void* d_out, int out_size, void* d_ws, size_t ws_size,
                              hipStream_t stream)
{
  (void)in_sizes; (void)n_in; (void)out_size; (void)ws_size;

  const float* x  = (const float*)d_in[0];
  const float* Wk = (const float*)d_in[1];
  const float* Wq = (const float*)d_in[2];
  const float* Wv = (const float*)d_in[3];
  const float* Wt = (const float*)d_in[4];
  float*       out = (float*)d_out;

  const size_t bufElems = (size_t)NPIX * CDIM;   // 2*64*16384 floats = 8 MB each
  float* Kbuf = (float*)d_ws;
  float* Qbuf = Kbuf + bufElems;
  float* Vbuf = Qbuf + bufElems;
  float* Abuf = Vbuf + bufElems;

  // 8 waves/block * 16 px/wave = 128 px per block; grid.y = 64/16 channel tiles
  dim3 ggrid(NPIX / 128, CDIM / 16);
  qkv_gemm_wmma<<<ggrid, 256, 0, stream>>>(Wk, Wq, Wv, x, Kbuf, Qbuf, Vbuf);

  attn49<<<NPIX / 256, 256, 0, stream>>>(Kbuf, Qbuf, Vbuf, Abuf);

  proj_gemm_wmma<<<ggrid, 256, 0, stream>>>(Wt, Abuf, out, x);
}